// PartialSum_85478439125876
// MI455X (gfx1250) — compile-verified
//
#include <hip/hip_runtime.h>

// PartialSum: out[b, p] = sum_{l=0..63} x[b, p*64 + l]
// BATCH=2048, NUM_PARTITIONS=1024, PARTITION_LENGTH=64 (fp32 in, fp32 out).
//
// Bandwidth-bound (512 MiB in / 8 MiB out => ~23 us floor at 23.3 TB/s).
// Reduction engine: chained V_WMMA_F32_16X16X4_F32 with B == ones.
// One wave32 owns a 16x64 fp32 tile (16 outputs, 4 KB contiguous input):
//   - 8 global_load_b128 (each float4 feeds two K=4 WMMA steps; valid because
//     B==ones makes the A-slot<->element mapping irrelevant to the sum)
//   - 2 independent 8-WMMA accumulation chains to hide D->C latency
//   - merge chains with 8 VALU adds, store 16 outputs as 2x global_store_b128

typedef float v2f __attribute__((ext_vector_type(2)));
typedef float v4f __attribute__((ext_vector_type(4)));
typedef float v8f __attribute__((ext_vector_type(8)));

#define PART_LEN        64   // reduction length per output
#define TILE_ROWS       16   // outputs per wave (WMMA M)
#define WAVES_PER_BLOCK 8    // 256 threads / wave32

__global__ __launch_bounds__(256, 4)
void PartialSum_wmma_kernel(const float* __restrict__ x,
                            float* __restrict__ out,
                            int num_tiles) {
    const int wave = blockIdx.x * WAVES_PER_BLOCK + (threadIdx.x >> 5);
    if (wave >= num_tiles) return;          // wave-uniform: EXEC stays all-1s

    const int lane = threadIdx.x & 31;
    const int m    = lane & 15;             // WMMA row (M) this lane feeds
    const int hi   = lane >> 4;             // lane half: elements [0,32) vs [32,64)

    // Lane (m,hi) streams floats row_m[hi*32 .. hi*32+31] as 8 float4 loads.
    // Across both halves each row's 64 elements are covered exactly once.
    const float* __restrict__ row =
        x + (size_t)wave * (TILE_ROWS * PART_LEN) + (size_t)m * PART_LEN + hi * 32;

    v8f c0 = {};                            // accumulation chain 0
    v8f c1 = {};                            // accumulation chain 1
    v2f b; b.x = 1.0f; b.y = 1.0f;          // B = ones: layout-invariant

    #pragma unroll
    for (int j = 0; j < 8; ++j) {
        v4f f = *(const v4f*)(row + j * 4); // global_load_b128, imm offset j*16B
        v2f a0; a0.x = f[0]; a0.y = f[1];
        v2f a1; a1.x = f[2]; a1.y = f[3];
        c0 = __builtin_amdgcn_wmma_f32_16x16x4_f32(
                 false, a0, false, b, (short)0, c0, false, false);
        c1 = __builtin_amdgcn_wmma_f32_16x16x4_f32(
                 false, a1, false, b, (short)0, c1, false, false);
    }

    // Merge the two chains: every column of (c0+c1) holds the 16 row sums.
    // D layout: VGPR r -> M=r (lanes 0-15, N=lane) / M=8+r (lanes 16-31).
    // Column N==0 lives in lane 0 (M=0..7) and lane 16 (M=8..15).
    if (m == 0) {
        float* o = out + (size_t)wave * TILE_ROWS + hi * 8;
        v4f lo4 = { c0[0] + c1[0], c0[1] + c1[1], c0[2] + c1[2], c0[3] + c1[3] };
        v4f hi4 = { c0[4] + c1[4], c0[5] + c1[5], c0[6] + c1[6], c0[7] + c1[7] };
        *(v4f*)(o)     = lo4;               // global_store_b128
        *(v4f*)(o + 4) = hi4;               // global_store_b128
    }
}

extern "C" void kernel_launch(void* const* d_in, const int* in_sizes, int n_in,
                              void* d_out, int out_size, void* d_ws, size_t ws_size,
                              hipStream_t stream) {
    const float* x = (const float*)d_in[0];
    float* out     = (float*)d_out;

    const int num_tiles = out_size / TILE_ROWS;                 // 131072 waves
    const int blocks    = (num_tiles + WAVES_PER_BLOCK - 1) / WAVES_PER_BLOCK;

    PartialSum_wmma_kernel<<<blocks, 256, 0, stream>>>(x, out, num_tiles);
}